// GraphSSD_80934363725835
// MI455X (gfx1250) — compile-verified
//
#include <hip/hip_runtime.h>
#include <hip/hip_bf16.h>
#include <math.h>

// ---------------------------------------------------------------------------
// GraphSSD on MI455X (gfx1250, wave32).
//   B_GRAPHS=8, L_NODES=512, H_HEADS=16, HEADDIM=64, D_STATE=64, E=65536, TOL=0.1
// Matrix inversion done as BLOCKED Gauss-Jordan (NB=64, no pivoting: G is
// strictly column diagonally dominant) so the trailing updates run on
// V_WMMA_F32_16X16X4_F32 instead of scalar VALU.
// ---------------------------------------------------------------------------

#define NG      8            // graphs
#define LN      512          // nodes per graph
#define NH      16           // heads
#define HD      64           // head dim
#define DS      64           // state dim
#define NE      65536        // edges
#define NN      (NG*LN)      // 4096 total nodes
#define NMAT    (NG*NH)      // 128 matrices (g,h)
#define MELEM   ((size_t)LN*LN)       // 262144 elems per 512x512 matrix
#define TOTM    ((size_t)NMAT*MELEM)  // 33,554,432 elems (128 MiB fp32)
#define NB      64                    // GJ block size
#define PEL     ((size_t)LN*NB)       // 32768 panel elems / matrix
#define DEL     ((size_t)NB*NB)       // 4096 diag-block elems / matrix

typedef __attribute__((ext_vector_type(2))) float v2f;
typedef __attribute__((ext_vector_type(8))) float v8f;

__device__ __forceinline__ float softplusf(float x) {
    return (x > 20.0f) ? x : log1pf(expf(x));
}

// ---------------------------------------------------------------------------
// 0) zero fill (scatter target must start at 0)
// ---------------------------------------------------------------------------
__global__ void k_zero(float* __restrict__ p, size_t n) {
    size_t i = (size_t)blockIdx.x * blockDim.x + threadIdx.x;
    if (i < n) p[i] = 0.0f;
}

// ---------------------------------------------------------------------------
// 1) prep: p0 = exp(0.5*dtA0), p1 = exp(0.5*dtA1), dummy = exp(dtA2)
// ---------------------------------------------------------------------------
__global__ void k_prep(const float* __restrict__ dt, const float* __restrict__ dtb,
                       float* __restrict__ p0, float* __restrict__ p1,
                       float* __restrict__ dummy) {
    int tid = blockIdx.x * blockDim.x + threadIdx.x;   // NN*NH = 65536
    int n = tid >> 4;
    int h = tid & 15;
    float bias = dtb[h];
    const float* dp = dt + (size_t)n * (NH*4) + h*4;
    float a0 = -softplusf(dp[0] + bias);
    float a1 = -softplusf(dp[1] + bias);
    float a2 = -softplusf(dp[2] + bias);
    p0[(size_t)h*NN + n] = expf(0.5f * a0);
    p1[(size_t)h*NN + n] = expf(0.5f * a1);
    int g = n >> 9, l = n & 511;
    dummy[((size_t)(g*NH + h))*LN + l] = expf(a2);
}

// ---------------------------------------------------------------------------
// 2) edge scatter: A[(g*NH+h)][sl][tl] += p0[h,src]*p1[h,dst]
// ---------------------------------------------------------------------------
__global__ void k_scatter(const int* __restrict__ eidx,
                          const float* __restrict__ p0, const float* __restrict__ p1,
                          float* __restrict__ A) {
    int tid = blockIdx.x * blockDim.x + threadIdx.x;   // NE*NH = 1,048,576
    int e = tid >> 4;
    int h = tid & 15;
    int src = eidx[e];
    int dst = eidx[NE + e];
    int g  = src >> 9;
    int sl = src & 511;
    int tl = dst & 511;
    float v = p0[(size_t)h*NN + src] * p1[(size_t)h*NN + dst];
    size_t off = (((size_t)(g*NH + h))*LN + sl)*LN + tl;
    atomicAdd(&A[off], v);
}

// ---------------------------------------------------------------------------
// 3) row sums: denom[mat*LN + i] = sum_j A[mat][i][j] + dummy + TOL
// ---------------------------------------------------------------------------
__global__ void k_rowsum(const float* __restrict__ A, const float* __restrict__ dummy,
                         float* __restrict__ denom) {
    int blk = blockIdx.x;              // NMAT*LN = 65536 blocks
    int mat = blk >> 9;
    int row = blk & 511;
    const float* ap = A + ((size_t)mat*LN + row)*LN;
    int t = threadIdx.x;               // 256
    __shared__ float red[256];
    red[t] = ap[t] + ap[t + 256];
    __syncthreads();
    for (int o = 128; o > 0; o >>= 1) {
        if (t < o) red[t] += red[t + o];
        __syncthreads();
    }
    if (t == 0) denom[blk] = red[0] + dummy[(size_t)mat*LN + row] + 0.1f;
}

// ---------------------------------------------------------------------------
// 4) build G = I - A_norm^T :  G[mat][j][i] = (i==j) - A[mat][i][j]/denom[mat,i]
// ---------------------------------------------------------------------------
__global__ void k_buildG(const float* __restrict__ A, const float* __restrict__ denom,
                         float* __restrict__ G) {
    size_t idx = (size_t)blockIdx.x * blockDim.x + threadIdx.x;   // TOTM
    if (idx >= TOTM) return;
    size_t mat = idx >> 18;
    int rem = (int)(idx & (MELEM - 1));
    int i = rem >> 9;
    int j = rem & 511;
    float an = A[idx] / denom[mat*LN + i];
    float gv = ((i == j) ? 1.0f : 0.0f) - an;
    G[(mat << 18) + (size_t)j*LN + i] = gv;
}

// ---------------------------------------------------------------------------
// 5) init V = I   (reuses A's buffer; A is no longer needed)
// ---------------------------------------------------------------------------
__global__ void k_initV(float* __restrict__ V) {
    size_t idx = (size_t)blockIdx.x * blockDim.x + threadIdx.x;
    if (idx >= TOTM) return;
    int rem = (int)(idx & (MELEM - 1));
    int i = rem >> 9;
    int j = rem & 511;
    V[idx] = (i == j) ? 1.0f : 0.0f;
}

// ---------------------------------------------------------------------------
// 6a) snapshot panel (transposed):  Pt[mat][j*LN + i] = G[mat][i][k0+j]
//     Transposed layout makes the update kernel's A-fragment loads
//     lane-contiguous.
// ---------------------------------------------------------------------------
__global__ void k_pcopy(const float* __restrict__ G, float* __restrict__ Pt, int k0) {
    size_t idx = (size_t)blockIdx.x * blockDim.x + threadIdx.x;  // NMAT*PEL = 4,194,304
    int mat = (int)(idx >> 15);
    int rem = (int)(idx & (PEL - 1));
    int j = rem >> 9;
    int i = rem & 511;
    Pt[idx] = G[(size_t)mat*MELEM + (size_t)i*LN + (k0 + j)];
}

// ---------------------------------------------------------------------------
// 6b) invert the 64x64 diagonal block D = G[S,S] in LDS (scalar GJ, tiny).
//     One 64-thread block per matrix; thread t owns column t.
// ---------------------------------------------------------------------------
__global__ void __launch_bounds__(64) k_dinv(const float* __restrict__ Pt,
                                             float* __restrict__ Dinv, int k0) {
    int mat = blockIdx.x;
    const float* Pm = Pt + (size_t)mat * PEL;
    int t = threadIdx.x;                        // 64
    __shared__ float sD[NB][NB + 1];
    __shared__ float sI[NB][NB + 1];
    __shared__ float cf[NB];
    for (int r = 0; r < NB; ++r) {
        // D[r][t] = G[k0+r][k0+t] = Pt[t*LN + k0 + r]
        sD[r][t] = Pm[(size_t)t*LN + (k0 + r)];
        sI[r][t] = (r == t) ? 1.0f : 0.0f;
    }
    __syncthreads();
    for (int k = 0; k < NB; ++k) {
        float ip = 1.0f / sD[k][k];
        float dk = sD[k][t] * ip;
        float ik = sI[k][t] * ip;
        __syncthreads();
        sD[k][t] = dk;
        sI[k][t] = ik;
        cf[t] = (t == k) ? 0.0f : sD[t][k];
        __syncthreads();
        for (int i = 0; i < NB; ++i) {
            float f = cf[i];
            sD[i][t] -= f * dk;
            sI[i][t] -= f * ik;
        }
        __syncthreads();
    }
    for (int r = 0; r < NB; ++r)
        Dinv[(size_t)mat*DEL + (size_t)r*NB + t] = sI[r][t];
}

// ---------------------------------------------------------------------------
// 6c) R' = Dinv x X[S,:]  for X in {G,V}.  WMMA GEMM, K = 64.
//     32768 waves exactly: ct(32) | rt(4) | tgt(2) | mat(128)
// ---------------------------------------------------------------------------
__global__ void k_rprime(const float* __restrict__ Dinv,
                         const float* __restrict__ G, const float* __restrict__ V,
                         float* __restrict__ RG, float* __restrict__ RV, int k0) {
    int w = (blockIdx.x * blockDim.x + threadIdx.x) >> 5;
    int lane = threadIdx.x & 31;
    int ct  = w & 31;
    int rt  = (w >> 5) & 3;
    int tgt = (w >> 7) & 1;
    int mat = w >> 8;
    const float* Dm = Dinv + (size_t)mat * DEL;
    const float* X  = (tgt ? V : G) + (size_t)mat * MELEM;
    float*       R  = (tgt ? RV : RG) + (size_t)mat * PEL;
    int half = lane >> 4;
    int l16  = lane & 15;
    int c    = ct*16 + l16;
    int arow = rt*16 + l16;
    v8f acc = {};
    #pragma unroll 4
    for (int kk = 0; kk < NB; kk += 4) {
        int ka = kk + half*2;
        v2f a, b;
        a.x = Dm[(size_t)arow*NB + ka];
        a.y = Dm[(size_t)arow*NB + ka + 1];
        b.x = X[(size_t)(k0 + ka    )*LN + c];
        b.y = X[(size_t)(k0 + ka + 1)*LN + c];
        acc = __builtin_amdgcn_wmma_f32_16x16x4_f32(false, a, false, b,
                                                    (short)0, acc, false, false);
    }
    #pragma unroll
    for (int r = 0; r < 8; ++r)
        R[(size_t)(rt*16 + r + half*8)*LN + c] = acc[r];
}

// ---------------------------------------------------------------------------
// 6d) trailing update:  rows outside S:  X_tile -= Panel(16x64) x R'(64x16)
//     rows inside  S:  X_tile = R' tile (the transformed pivot rows)
//     Both branches are wave-uniform.  A-fragment negated in VALU (f32 WMMA
//     has no A/B NEG per ISA).  nct/ct0 shrink the G update each step since
//     columns < k0 of G are provably zero-producing.
// ---------------------------------------------------------------------------
__global__ void k_update(float* __restrict__ X, const float* __restrict__ Rt,
                         const float* __restrict__ Pt, int k0, int ct0, int nct) {
    int w = (blockIdx.x * blockDim.x + threadIdx.x) >> 5;
    int lane = threadIdx.x & 31;
    int ct  = ct0 + (w % nct);
    int tmp = w / nct;
    int rt  = tmp & 31;
    int mat = tmp >> 5;
    float*       Xm = X  + (size_t)mat * MELEM;
    const float* Rm = Rt + (size_t)mat * PEL;
    int half = lane >> 4;
    int l16  = lane & 15;
    int c    = ct*16 + l16;
    int rS   = k0 >> 4;
    if (rt >= rS && rt < rS + 4) {
        // pivot-row tile: becomes R'
        #pragma unroll
        for (int r = 0; r < 8; ++r) {
            int row = rt*16 + r + half*8;
            Xm[(size_t)row*LN + c] = Rm[(size_t)(row - k0)*LN + c];
        }
    } else {
        const float* Pm = Pt + (size_t)mat * PEL;   // transposed panel [j*LN + i]
        int r0 = rt*16;
        v8f acc;
        #pragma unroll
        for (int r = 0; r < 8; ++r)
            acc[r] = Xm[(size_t)(r0 + r + half*8)*LN + c];
        #pragma unroll 4
        for (int kk = 0; kk < NB; kk += 4) {
            int ka = kk + half*2;
            v2f a, b;
            a.x = -Pm[(size_t) ka     *LN + r0 + l16];
            a.y = -Pm[(size_t)(ka + 1)*LN + r0 + l16];
            b.x =  Rm[(size_t) ka     *LN + c];
            b.y =  Rm[(size_t)(ka + 1)*LN + c];
            acc = __builtin_amdgcn_wmma_f32_16x16x4_f32(false, a, false, b,
                                                        (short)0, acc, false, false);
        }
        #pragma unroll
        for (int r = 0; r < 8; ++r)
            Xm[(size_t)(r0 + r + half*8)*LN + c] = acc[r];
    }
}

// ---------------------------------------------------------------------------
// 7) CB[g][l][t] = sum_d C[g,l,d] * B[g,t,d]   via WMMA, K = 64.
// ---------------------------------------------------------------------------
__global__ void k_cb_wmma(const float* __restrict__ C, const float* __restrict__ Bm,
                          float* __restrict__ CB) {
    int wave = (blockIdx.x * blockDim.x + threadIdx.x) >> 5;  // 8192 waves exactly
    int lane = threadIdx.x & 31;
    int tj = wave & 31;
    int ti = (wave >> 5) & 31;
    int g  = wave >> 10;
    int i0 = ti * 16, j0 = tj * 16;
    const float* Cg = C  + (size_t)g * LN * DS;
    const float* Bg = Bm + (size_t)g * LN * DS;
    int half = lane >> 4;
    int l16  = lane & 15;
    v8f acc = {};
    #pragma unroll 4
    for (int k0 = 0; k0 < DS; k0 += 4) {
        int ka = k0 + half * 2;
        const float* cp = Cg + (size_t)(i0 + l16) * DS + ka;
        const float* bp = Bg + (size_t)(j0 + l16) * DS + ka;
        v2f a, b;
        a.x = cp[0];  a.y = cp[1];
        b.x = bp[0];  b.y = bp[1];
        acc = __builtin_amdgcn_wmma_f32_16x16x4_f32(false, a, false, b,
                                                    (short)0, acc, false, false);
    }
    float* out = CB + (size_t)g * MELEM;
    int col = j0 + l16;
    #pragma unroll
    for (int r = 0; r < 8; ++r) {
        int row = i0 + r + half * 8;
        out[(size_t)row * LN + col] = acc[r];
    }
}

// ---------------------------------------------------------------------------
// 8) y = (Linv o CB o dt_self-cols) @ xr + D[h]*xr, written as (b,L,hd,H).
// ---------------------------------------------------------------------------
__global__ void k_y_wmma(const float* __restrict__ x, const float* __restrict__ dt,
                         const float* __restrict__ Dv,
                         const float* __restrict__ Vinv, const float* __restrict__ CB,
                         float* __restrict__ out) {
    int wave = (blockIdx.x * blockDim.x + threadIdx.x) >> 5;  // 16384 waves exactly
    int lane = threadIdx.x & 31;
    int tc  = wave & 3;            // hd tile (4)
    int tl  = (wave >> 2) & 31;    // row tile (32)
    int mat = wave >> 7;           // (g,h) 0..127
    int g = mat >> 4;
    int h = mat & 15;
    int l0 = tl * 16, c0 = tc * 16;
    const float* Vm = Vinv + (size_t)mat * MELEM;
    const float* cb = CB   + (size_t)g   * MELEM;
    int half = lane >> 4;
    int l16  = lane & 15;
    int rowm = l0 + l16;
    const float* vrow = Vm + (size_t)rowm * LN;
    const float* crow = cb + (size_t)rowm * LN;
    v8f acc = {};
    for (int t0 = 0; t0 < LN; t0 += 4) {
        int ta = t0 + half * 2;
        v2f a, b;
        a.x = vrow[ta]     * crow[ta];
        a.y = vrow[ta + 1] * crow[ta + 1];
        size_t n0 = (size_t)(g * LN + ta);
        const float* xp = x + n0 * (NH*HD) + h * HD + c0 + l16;
        float s0 = dt[ n0      * (NH*4) + h*4 + 3];
        float s1 = dt[(n0 + 1) * (NH*4) + h*4 + 3];
        b.x = s0 * xp[0];
        b.y = s1 * xp[NH*HD];
        acc = __builtin_amdgcn_wmma_f32_16x16x4_f32(false, a, false, b,
                                                    (short)0, acc, false, false);
    }
    float dcoef = Dv[h];
    int cdim = c0 + l16;
    #pragma unroll
    for (int r = 0; r < 8; ++r) {
        int l = l0 + r + half * 8;
        size_t n = (size_t)(g * LN + l);
        float xv = x[n * (NH*HD) + h * HD + cdim];
        out[n * (NH*HD) + (size_t)cdim * NH + h] = acc[r] + dcoef * xv;
    }
}

// ---------------------------------------------------------------------------
// launcher
// ---------------------------------------------------------------------------
extern "C" void kernel_launch(void* const* d_in, const int* in_sizes, int n_in,
                              void* d_out, int out_size, void* d_ws, size_t ws_size,
                              hipStream_t stream) {
    const float* x   = (const float*)d_in[0];
    const float* Bm  = (const float*)d_in[1];
    const float* Cm  = (const float*)d_in[2];
    const float* dt  = (const float*)d_in[3];
    const float* dtb = (const float*)d_in[4];
    const float* Dv  = (const float*)d_in[5];
    const int*  eidx = (const int*)d_in[6];
    float* out = (float*)d_out;

    // workspace layout (floats)
    float* ws    = (float*)d_ws;
    float* Abuf  = ws;                           // scatter target -> V -> Linv
    float* Gbuf  = Abuf + TOTM;                  // I - A_norm^T
    float* CBbuf = Gbuf + TOTM;                  // NG*LN*LN
    float* p0    = CBbuf + (size_t)NG * MELEM;
    float* p1    = p0 + (size_t)NH * NN;
    float* dummy = p1 + (size_t)NH * NN;
    float* denom = dummy + (size_t)NMAT * LN;
    float* Pbuf  = denom + (size_t)NMAT * LN;    // NMAT*PEL  (panel, transposed)
    float* Dbuf  = Pbuf + (size_t)NMAT * PEL;    // NMAT*DEL  (D^-1)
    float* RG    = Dbuf + (size_t)NMAT * DEL;    // NMAT*PEL
    float* RV    = RG   + (size_t)NMAT * PEL;    // NMAT*PEL

    const int T = 256;
    unsigned gTot = (unsigned)((TOTM + T - 1) / T);          // 131072 blocks

    k_zero   <<<gTot, T, 0, stream>>>(Abuf, TOTM);
    k_prep   <<<(NN*NH)/T, T, 0, stream>>>(dt, dtb, p0, p1, dummy);
    k_scatter<<<(NE*NH)/T, T, 0, stream>>>(eidx, p0, p1, Abuf);
    k_rowsum <<<NMAT*LN, T, 0, stream>>>(Abuf, dummy, denom);
    k_buildG <<<gTot, T, 0, stream>>>(Abuf, denom, Gbuf);
    k_initV  <<<gTot, T, 0, stream>>>(Abuf);                 // Abuf becomes V

    // blocked Gauss-Jordan: 8 block steps of NB=64
    for (int p = 0; p < LN / NB; ++p) {
        int k0 = p * NB;
        k_pcopy <<<(unsigned)((NMAT*PEL)/T), T, 0, stream>>>(Gbuf, Pbuf, k0);
        k_dinv  <<<NMAT, NB, 0, stream>>>(Pbuf, Dbuf, k0);
        k_rprime<<<(NMAT*2*4*32*32)/T, T, 0, stream>>>(Dbuf, Gbuf, Abuf, RG, RV, k0);
        int nctG = 32 - p*4;                                  // skip G cols < k0
        k_update<<<(NMAT*32*nctG*32)/T, T, 0, stream>>>(Gbuf, RG, Pbuf, k0, p*4, nctG);
        k_update<<<(NMAT*32*32*32)/T, T, 0, stream>>>(Abuf, RV, Pbuf, k0, 0, 32);
    }
    // Abuf now holds Linv = (I - A_norm^T)^-1

    k_cb_wmma<<<(8192*32)/T, T, 0, stream>>>(Cm, Bm, CBbuf);
    k_y_wmma <<<(16384*32)/T, T, 0, stream>>>(x, dt, Dv, Abuf, CBbuf, out);
}